// NegativeLogLikelihood_90469191123509
// MI455X (gfx1250) — compile-verified
//
#include <hip/hip_runtime.h>

typedef __attribute__((ext_vector_type(2))) float v2f;
typedef __attribute__((ext_vector_type(8))) float v8f;

#define TPB   256
#define ITEMS 16
#define EPB   (TPB * ITEMS)   // 4096 elements per block

// ---------- WMMA-based wave reduction (wave32, EXEC must be all 1s) ----------
// A (16x4 f32): lane L<16 holds A[L,0],A[L,1]; lane L>=16 holds A[L-16,2],A[L-16,3].
// With a = {x, 0} per lane and B = ones, D[m,n] = x_m + x_{m+16}.
// D (16x16 f32): VGPR r, lanes 0-15 -> M=r, lanes 16-31 -> M=r+8.
// Sum the 8 D VGPRs per lane, then fold halves with shfl_xor(16) -> full 32-lane sum.
__device__ __forceinline__ float wave_reduce_wmma(float x) {
    v2f a; a[0] = x;    a[1] = 0.0f;
    v2f b; b[0] = 1.0f; b[1] = 1.0f;
    v8f c = {};
    v8f d = __builtin_amdgcn_wmma_f32_16x16x4_f32(
        /*neg_a=*/false, a, /*neg_b=*/false, b,
        /*c_mod=*/(short)0, c, /*reuse_a=*/false, /*reuse_b=*/false);
    float s = ((d[0] + d[1]) + (d[2] + d[3])) + ((d[4] + d[5]) + (d[6] + d[7]));
    s += __shfl_xor(s, 16, 32);
    return s;
}

// Block reduction: WMMA per wave, then tiny serial combine (fixed order).
__device__ __forceinline__ float block_reduce_wmma(float x, float* lds8) {
    const int tid = threadIdx.x;
    __syncthreads();                       // protect lds8 reuse between calls
    float w = wave_reduce_wmma(x);         // all lanes active here
    if ((tid & 31) == 0) lds8[tid >> 5] = w;
    __syncthreads();
    if (tid == 0) {
        float s = 0.0f;
        #pragma unroll
        for (int i = 0; i < TPB / 32; ++i) s += lds8[i];
        lds8[0] = s;
    }
    __syncthreads();
    return lds8[0];
}

// Kogge-Stone inclusive scan of one double per thread across the block.
__device__ __forceinline__ double block_incl_scan(double v, double* sc) {
    const int tid = threadIdx.x;
    sc[tid] = v;
    __syncthreads();
    #pragma unroll
    for (int off = 1; off < TPB; off <<= 1) {
        double mine  = sc[tid];
        double other = (tid >= off) ? sc[tid - off] : 0.0;
        __syncthreads();
        sc[tid] = mine + other;
        __syncthreads();
    }
    double r = sc[tid];
    __syncthreads();
    return r;
}

// ---------------- K1: per-block sum of exp(pred) ----------------
__global__ __launch_bounds__(TPB)
void k1_block_sums(const float* __restrict__ pred, float* __restrict__ blockExp, int n) {
    __shared__ float lds8[8];
    const long base = (long)blockIdx.x * EPB + (long)threadIdx.x * ITEMS;
    float s = 0.0f;
    if (base + ITEMS <= n) {
        const float4* p4 = reinterpret_cast<const float4*>(pred + base);
        #pragma unroll
        for (int v = 0; v < ITEMS / 4; ++v) {
            float4 f = p4[v];
            s += (__expf(f.x) + __expf(f.y)) + (__expf(f.z) + __expf(f.w));
        }
    } else {
        for (int j = 0; j < ITEMS; ++j) {
            long i = base + j;
            if (i < n) s += __expf(pred[i]);
        }
    }
    float tot = block_reduce_wmma(s, lds8);
    if (threadIdx.x == 0) blockExp[blockIdx.x] = tot;
}

// ---------------- K2: exclusive scan of block sums (double), one block ----------------
__global__ __launch_bounds__(TPB)
void k2_scan_offsets(const float* __restrict__ blockExp, double* __restrict__ blockOff,
                     int nblocks) {
    __shared__ double sc[TPB];
    const int tid  = threadIdx.x;
    const int base = tid * ITEMS;
    double excl[ITEMS];
    double run = 0.0;
    #pragma unroll
    for (int j = 0; j < ITEMS; ++j) {
        int i = base + j;
        double v = (i < nblocks) ? (double)blockExp[i] : 0.0;
        excl[j] = run;          // exclusive within this thread
        run += v;
    }
    double incl  = block_incl_scan(run, sc);
    double texcl = incl - run;  // exclusive prefix of this thread's chunk
    #pragma unroll
    for (int j = 0; j < ITEMS; ++j) {
        int i = base + j;
        if (i < nblocks) blockOff[i] = texcl + excl[j];
    }
}

// ---------------- K3: per-block dot = sum(ev * (pred - log(cumsum))) ----------------
__global__ __launch_bounds__(TPB)
void k3_dot(const float* __restrict__ pred, const int* __restrict__ yev,
            const double* __restrict__ blockOff,
            float* __restrict__ dotPart, float* __restrict__ evPart, int n) {
    __shared__ double sc[TPB];
    __shared__ float  lds8[8];
    const int  tid  = threadIdx.x;
    const long base = (long)blockIdx.x * EPB + (long)tid * ITEMS;

    float p[ITEMS]; int e[ITEMS]; float ex[ITEMS];
    if (base + ITEMS <= n) {
        const float4* p4 = reinterpret_cast<const float4*>(pred + base);
        const int4*   e4 = reinterpret_cast<const int4*>(yev + base);
        #pragma unroll
        for (int v = 0; v < ITEMS / 4; ++v) {
            float4 f = p4[v]; int4 g = e4[v];
            p[4*v+0] = f.x; p[4*v+1] = f.y; p[4*v+2] = f.z; p[4*v+3] = f.w;
            e[4*v+0] = g.x; e[4*v+1] = g.y; e[4*v+2] = g.z; e[4*v+3] = g.w;
        }
        #pragma unroll
        for (int j = 0; j < ITEMS; ++j) ex[j] = __expf(p[j]);
    } else {
        for (int j = 0; j < ITEMS; ++j) {
            long i = base + j;
            bool ok = (i < n);
            p[j]  = ok ? pred[i] : 0.0f;
            e[j]  = ok ? yev[i]  : 0;
            ex[j] = ok ? __expf(p[j]) : 0.0f;
        }
    }

    float run = 0.0f; float lrun[ITEMS];
    #pragma unroll
    for (int j = 0; j < ITEMS; ++j) { run += ex[j]; lrun[j] = run; }

    double incl = block_incl_scan((double)run, sc);
    double off0 = blockOff[blockIdx.x] + (incl - (double)run);

    double dot = 0.0; int ec = 0;
    #pragma unroll
    for (int j = 0; j < ITEMS; ++j) {
        if (e[j] != 0) {
            double S = off0 + (double)lrun[j];      // inclusive cumsum at element j
            dot += (double)e[j] * (double)(p[j] - __logf((float)S));
            ec  += e[j];
        }
    }

    float bdot = block_reduce_wmma((float)dot, lds8);
    float bev  = block_reduce_wmma((float)ec,  lds8);
    if (tid == 0) { dotPart[blockIdx.x] = bdot; evPart[blockIdx.x] = bev; }
}

// ---------------- K4: final reduction, write scalar cost ----------------
__global__ __launch_bounds__(TPB)
void k4_final(const float* __restrict__ dotPart, const float* __restrict__ evPart,
              float* __restrict__ out, int nblocks) {
    __shared__ float lds8[8];
    const int tid = threadIdx.x;
    double d = 0.0, ev = 0.0;
    for (int i = tid; i < nblocks; i += TPB) {
        d  += (double)dotPart[i];
        ev += (double)evPart[i];
    }
    float bd = block_reduce_wmma((float)d,  lds8);
    float be = block_reduce_wmma((float)ev, lds8);
    if (tid == 0) out[0] = -(bd / be);
}

extern "C" void kernel_launch(void* const* d_in, const int* in_sizes, int n_in,
                              void* d_out, int out_size, void* d_ws, size_t ws_size,
                              hipStream_t stream) {
    const float* pred = (const float*)d_in[0];
    // d_in[1] (ytime) defines the ordering only; unused by the math.
    const int*   yev  = (const int*)d_in[2];
    const int n = in_sizes[0];
    const int nblocks = (n + EPB - 1) / EPB;   // 4096 for N = 2^24

    char*   ws       = (char*)d_ws;
    float*  blockExp = (float*)(ws);                 // 4096 * 4  = 16 KB
    double* blockOff = (double*)(ws + 16384);        // 4096 * 8  = 32 KB
    float*  dotPart  = (float*)(ws + 49152);         // 4096 * 4  = 16 KB
    float*  evPart   = (float*)(ws + 65536);         // 4096 * 4  = 16 KB

    k1_block_sums <<<nblocks, TPB, 0, stream>>>(pred, blockExp, n);
    k2_scan_offsets<<<1,       TPB, 0, stream>>>(blockExp, blockOff, nblocks);
    k3_dot        <<<nblocks, TPB, 0, stream>>>(pred, yev, blockOff, dotPart, evPart, n);
    k4_final      <<<1,       TPB, 0, stream>>>(dotPart, evPart, (float*)d_out, nblocks);
}